// GroupPatientLearnerModule_30605936951451
// MI455X (gfx1250) — compile-verified
//
#include <hip/hip_runtime.h>
#include <math.h>

// ---------------------------------------------------------------------------
// GroupPatientLearner fused CDNA5 implementation.
//
// Key insight: emb[:B] is dead code in the reference. Only
//   adj[:B, B:]   (4096 x 256)  -> argmax            (group_label)
//   adj[B:, :]    (256 x 4352)  -> @ g, MLP, sigmoid (group_embedding)
// are needed, so we never materialize the 4352^2 adjacency.
//
// Gram tiles (X.X^T and E.E^T, K = 19 padded to 20) and the adj@g product run
// on V_WMMA_F32_16X16X4_F32 (f32-exact, matches reference numerics).
// ---------------------------------------------------------------------------

typedef float v2f __attribute__((ext_vector_type(2)));
typedef float v8f __attribute__((ext_vector_type(8)));

#define B_   4096
#define D_   19
#define F_   64
#define NC_  256
#define HID_ 32
#define SQ_  16
#define N_   (B_ + NC_)   // 4352
#define KP_  20           // feature dim padded to multiple of 4 (zeros in col 19)

// d_out layout (floats, concatenated in tuple return order)
#define OUT_SCORES 0
#define OUT_GE     (B_ * D_)              // 77824
#define OUT_GPE    (OUT_GE + NC_ * D_)    // 82688
#define OUT_LABEL  (OUT_GPE + B_ * D_)    // 160512 ; total 164608

// d_ws layout (floats); total = 331520 floats = ~1.27 MB
#define WS_XALL 0
#define WS_EALL (WS_XALL + N_ * KP_)      // 87040
#define WS_SQ   (WS_EALL + N_ * KP_)      // 174080
#define WS_G    (WS_SQ + N_)              // 178432
#define WS_GE   (WS_G + N_ * HID_)        // 317696
#define WS_SC   (WS_GE + NC_ * D_)        // 322560
#define WS_LBL  (WS_SC + NC_ * D_)        // 327424 (int32), + 4096

// f32 16x4 A / 4x16 B operand loader for V_WMMA_F32_16X16X4_F32:
// lane L supplies row base+(L&15), columns k0 + 2*(L>>4) + {0,1}.
__device__ __forceinline__ v2f load_pair20(const float* __restrict__ p,
                                           int rowbase, int lane, int k0) {
  int row = rowbase + (lane & 15);
  int k   = k0 + ((lane >> 4) << 1);
  const float* q = p + row * KP_ + k;
  v2f r = { q[0], q[1] };
  return r;
}

// ---------------------------------------------------------------------------
// Kernel 1a: build x_all (x.mean over F ++ centers@proj_w+proj_b) and
//            e_all = exp(1 - dc_all), both padded N x 20 (pad col = 0).
// ---------------------------------------------------------------------------
__global__ void k_prep(const float* __restrict__ x, const float* __restrict__ dc,
                       const float* __restrict__ centers,
                       const float* __restrict__ proj_w, const float* __restrict__ proj_b,
                       const float* __restrict__ cc, float* __restrict__ ws) {
  int tid = blockIdx.x * blockDim.x + threadIdx.x;
  if (tid >= N_ * KP_) return;
  int n = tid / KP_, d = tid % KP_;
  float xv = 0.0f, ev = 0.0f;
  if (d < D_) {
    if (n < B_) {
      const float* px = x + ((size_t)n * D_ + d) * F_;
      float s = 0.0f;
#pragma unroll 8
      for (int f = 0; f < F_; ++f) s += px[f];
      xv = s * (1.0f / (float)F_);
      ev = expf(1.0f - dc[n * D_ + d]);
    } else {
      int c = n - B_;
      float s = proj_b[d];
      for (int k = 0; k < 61; ++k) s += centers[c * 61 + k] * proj_w[k * D_ + d];
      xv = s;
      ev = expf(1.0f - cc[c * D_ + d]);
    }
  }
  ws[WS_XALL + n * KP_ + d] = xv;
  ws[WS_EALL + n * KP_ + d] = ev;
}

// ---------------------------------------------------------------------------
// Kernel 1b: g = x_all @ gcn_w  (N x 32) and sq[n] = sum_d x_all[n][d]^2.
// ---------------------------------------------------------------------------
__global__ void k_prep2(const float* __restrict__ gcn_w, float* __restrict__ ws) {
  int tid = blockIdx.x * blockDim.x + threadIdx.x;
  if (tid >= N_ * HID_) return;
  int n = tid / HID_, h = tid % HID_;
  const float* xr = ws + WS_XALL + n * KP_;
  float s = 0.0f;
#pragma unroll
  for (int d = 0; d < D_; ++d) s += xr[d] * gcn_w[d * HID_ + h];
  ws[WS_G + n * HID_ + h] = s;
  if (h == 0) {
    float q = 0.0f;
#pragma unroll
    for (int d = 0; d < D_; ++d) q += xr[d] * xr[d];
    ws[WS_SQ + n] = q;
  }
}

// ---------------------------------------------------------------------------
// Kernel 2a: group_label = argmax_j adj[:B, B:].  One wave per 16 patient rows;
// 16 WMMA-built adjacency tiles per workgroup, argmax via LDS row scan.
// ---------------------------------------------------------------------------
__global__ __launch_bounds__(32)
void k_labels(const float* __restrict__ ws, const float* __restrict__ dcp,
              float* __restrict__ out, int* __restrict__ label_ws) {
  const int lane = threadIdx.x;
  const int hi = lane >> 4;
  const int lo = lane & 15;
  const int i_base = blockIdx.x * 16;
  const float* xall = ws + WS_XALL;
  const float* eall = ws + WS_EALL;
  const float* sq   = ws + WS_SQ;
  const float p = dcp[0];
  const float inv_d = 1.0f / (float)D_;

  v2f ax[5], ae[5];
#pragma unroll
  for (int s = 0; s < 5; ++s) {
    ax[s] = load_pair20(xall, i_base, lane, 4 * s);
    ae[s] = load_pair20(eall, i_base, lane, 4 * s);
  }
  float si[8];
#pragma unroll
  for (int r = 0; r < 8; ++r) si[r] = sq[i_base + r + 8 * hi];

  __shared__ float tile[256];
  float best = -INFINITY;
  int bestj = 0;

  for (int t = 0; t < NC_ / 16; ++t) {
    const int j_base = B_ + t * 16;
    const float sqj = sq[j_base + lo];
    v8f cx = {};
    v8f ce = {};
#pragma unroll
    for (int s = 0; s < 5; ++s) {
      v2f bx = load_pair20(xall, j_base, lane, 4 * s);
      v2f be = load_pair20(eall, j_base, lane, 4 * s);
      cx = __builtin_amdgcn_wmma_f32_16x16x4_f32(false, ax[s], false, bx,
                                                 (short)0, cx, false, false);
      ce = __builtin_amdgcn_wmma_f32_16x16x4_f32(false, ae[s], false, be,
                                                 (short)0, ce, false, false);
    }
#pragma unroll
    for (int r = 0; r < 8; ++r) {
      float sqdist = si[r] + sqj - 2.0f * cx[r];
      float denom = ((1.0f - p) * sqdist + p * ce[r]) * inv_d;
      tile[(r + 8 * hi) * 16 + lo] = 1.0f / denom;   // i<B, j>=B: never diagonal
    }
    __syncthreads();
    if (lane < 16) {
#pragma unroll
      for (int jj = 0; jj < 16; ++jj) {
        float v = tile[lane * 16 + jj];
        if (v > best) { best = v; bestj = t * 16 + jj; }  // strict > keeps first max
      }
    }
    __syncthreads();
  }
  if (lane < 16) {
    int i = i_base + lane;
    label_ws[i] = bestj;
    out[OUT_LABEL + i] = (float)bestj;
  }
}

// ---------------------------------------------------------------------------
// Kernel 2b: center rows.  emb[B:] = sigmoid((adj[B:,:] @ g + gcn_b) @ out_w + out_b).
// Per 16 center rows: loop all 272 j-tiles; WMMA gram -> adj tile (VALU) ->
// LDS transpose -> WMMA adj@g accumulation into a 16x32 accumulator.
// ---------------------------------------------------------------------------
__global__ __launch_bounds__(32)
void k_center_emb(const float* __restrict__ ws, const float* __restrict__ dcp,
                  const float* __restrict__ gcn_b, const float* __restrict__ out_w,
                  const float* __restrict__ out_b,
                  float* __restrict__ ge_ws, float* __restrict__ out) {
  const int lane = threadIdx.x;
  const int hi = lane >> 4;
  const int lo = lane & 15;
  const int i_base = B_ + blockIdx.x * 16;
  const float* xall = ws + WS_XALL;
  const float* eall = ws + WS_EALL;
  const float* sq   = ws + WS_SQ;
  const float* g    = ws + WS_G;
  const float p = dcp[0];
  const float inv_d = 1.0f / (float)D_;

  v2f ax[5], ae[5];
#pragma unroll
  for (int s = 0; s < 5; ++s) {
    ax[s] = load_pair20(xall, i_base, lane, 4 * s);
    ae[s] = load_pair20(eall, i_base, lane, 4 * s);
  }
  float si[8];
#pragma unroll
  for (int r = 0; r < 8; ++r) si[r] = sq[i_base + r + 8 * hi];

  __shared__ float tile[256];          // 16x16 adjacency tile (row-major)
  __shared__ float embLds[16 * HID_];  // 16x32 emb tile for the out-proj

  v8f embA = {};  // columns h = 0..15
  v8f embB = {};  // columns h = 16..31

  for (int t = 0; t < N_ / 16; ++t) {
    const int j_base = t * 16;
    __builtin_prefetch(xall + (j_base + 16) * KP_, 0, 1);  // global_prefetch_b8
    const float sqj = sq[j_base + lo];
    v8f cx = {};
    v8f ce = {};
#pragma unroll
    for (int s = 0; s < 5; ++s) {
      v2f bx = load_pair20(xall, j_base, lane, 4 * s);
      v2f be = load_pair20(eall, j_base, lane, 4 * s);
      cx = __builtin_amdgcn_wmma_f32_16x16x4_f32(false, ax[s], false, bx,
                                                 (short)0, cx, false, false);
      ce = __builtin_amdgcn_wmma_f32_16x16x4_f32(false, ae[s], false, be,
                                                 (short)0, ce, false, false);
    }
#pragma unroll
    for (int r = 0; r < 8; ++r) {
      int gi = i_base + r + 8 * hi;
      int gj = j_base + lo;
      float sqdist = si[r] + sqj - 2.0f * cx[r];
      float denom = ((1.0f - p) * sqdist + p * ce[r]) * inv_d;
      float v = 1.0f / denom;
      if (gi == gj) v = 1.0f;          // sim*(1-eye)+eye
      tile[(r + 8 * hi) * 16 + lo] = v;
    }
    __syncthreads();
    // adj(16x16) @ g(16x32): K = j_local, 4 k-steps.
    // A lane L: adj[L&15][k0 + 2*(L>>4) + v]; B lane L: g[j_base+k][ (L&15) (+16) ].
#pragma unroll
    for (int k0 = 0; k0 < 16; k0 += 4) {
      int kk = k0 + 2 * hi;
      v2f aA = { tile[lo * 16 + kk], tile[lo * 16 + kk + 1] };
      const float* g0 = g + (size_t)(j_base + kk) * HID_;
      const float* g1 = g0 + HID_;
      v2f b0 = { g0[lo], g1[lo] };
      v2f b1 = { g0[16 + lo], g1[16 + lo] };
      embA = __builtin_amdgcn_wmma_f32_16x16x4_f32(false, aA, false, b0,
                                                   (short)0, embA, false, false);
      embB = __builtin_amdgcn_wmma_f32_16x16x4_f32(false, aA, false, b1,
                                                   (short)0, embB, false, false);
    }
    __syncthreads();
  }

  // + gcn_b, stash 16x32 emb tile in LDS
  const float gb0 = gcn_b[lo];
  const float gb1 = gcn_b[16 + lo];
#pragma unroll
  for (int r = 0; r < 8; ++r) {
    int il = r + 8 * hi;
    embLds[il * HID_ + lo]      = embA[r] + gb0;
    embLds[il * HID_ + 16 + lo] = embB[r] + gb1;
  }
  __syncthreads();

  // group_embedding = sigmoid(emb @ out_w + out_b), 16 rows x 19 cols
  for (int idx = lane; idx < 16 * D_; idx += 32) {
    int row = idx / D_;
    int d   = idx % D_;
    float s = out_b[d];
#pragma unroll
    for (int h = 0; h < HID_; ++h) s += embLds[row * HID_ + h] * out_w[h * D_ + d];
    float v = 1.0f / (1.0f + expf(-s));
    int gc = (i_base - B_) + row;      // 0..255
    ge_ws[gc * D_ + d] = v;
    out[OUT_GE + gc * D_ + d] = v;
  }
}

// ---------------------------------------------------------------------------
// Kernel 3: scores_per_group = sigmoid(gelu_exact(ge @ imp_w1 + b1) @ imp_w2 + b2)
// ---------------------------------------------------------------------------
__global__ void k_imp(const float* __restrict__ imp_w1, const float* __restrict__ imp_b1,
                      const float* __restrict__ imp_w2, const float* __restrict__ imp_b2,
                      float* __restrict__ ws) {
  int row = blockIdx.x * blockDim.x + threadIdx.x;
  if (row >= NC_) return;
  const float* ge = ws + WS_GE + row * D_;
  float h[SQ_];
#pragma unroll
  for (int k = 0; k < SQ_; ++k) {
    float s = imp_b1[k];
    for (int d = 0; d < D_; ++d) s += ge[d] * imp_w1[d * SQ_ + k];
    h[k] = 0.5f * s * (1.0f + erff(s * 0.7071067811865475f));  // exact GELU
  }
#pragma unroll
  for (int d = 0; d < D_; ++d) {
    float s = imp_b2[d];
    for (int k = 0; k < SQ_; ++k) s += h[k] * imp_w2[k * D_ + d];
    ws[WS_SC + row * D_ + d] = 1.0f / (1.0f + expf(-s));
  }
}

// ---------------------------------------------------------------------------
// Kernel 4: gather per-patient outputs via group_label.
// ---------------------------------------------------------------------------
__global__ void k_gather(const float* __restrict__ ws, const int* __restrict__ label_ws,
                         float* __restrict__ out) {
  int i = blockIdx.x * blockDim.x + threadIdx.x;
  if (i >= B_) return;
  int lbl = label_ws[i];
  const float* sc = ws + WS_SC + lbl * D_;
  const float* ge = ws + WS_GE + lbl * D_;
#pragma unroll
  for (int d = 0; d < D_; ++d) {
    out[OUT_SCORES + i * D_ + d] = sc[d];
    out[OUT_GPE    + i * D_ + d] = ge[d];
  }
}

// ---------------------------------------------------------------------------
extern "C" void kernel_launch(void* const* d_in, const int* in_sizes, int n_in,
                              void* d_out, int out_size, void* d_ws, size_t ws_size,
                              hipStream_t stream) {
  const float* x       = (const float*)d_in[0];
  const float* dc      = (const float*)d_in[1];
  const float* centers = (const float*)d_in[2];
  const float* proj_w  = (const float*)d_in[3];
  const float* proj_b  = (const float*)d_in[4];
  const float* dcp     = (const float*)d_in[5];
  const float* cc      = (const float*)d_in[6];
  const float* gcn_w   = (const float*)d_in[7];
  const float* gcn_b   = (const float*)d_in[8];
  const float* out_w   = (const float*)d_in[9];
  const float* out_b   = (const float*)d_in[10];
  const float* imp_w1  = (const float*)d_in[11];
  const float* imp_b1  = (const float*)d_in[12];
  const float* imp_w2  = (const float*)d_in[13];
  const float* imp_b2  = (const float*)d_in[14];
  (void)in_sizes; (void)n_in; (void)out_size; (void)ws_size;

  float* out = (float*)d_out;
  float* ws  = (float*)d_ws;               // needs ~1.27 MB
  int*   label_ws = (int*)(ws + WS_LBL);
  float* ge_ws    = ws + WS_GE;

  k_prep<<<(N_ * KP_ + 255) / 256, 256, 0, stream>>>(x, dc, centers, proj_w, proj_b, cc, ws);
  k_prep2<<<(N_ * HID_ + 255) / 256, 256, 0, stream>>>(gcn_w, ws);
  k_labels<<<B_ / 16, 32, 0, stream>>>(ws, dcp, out, label_ws);
  k_center_emb<<<NC_ / 16, 32, 0, stream>>>(ws, dcp, gcn_b, out_w, out_b, ge_ws, out);
  k_imp<<<1, 256, 0, stream>>>(imp_w1, imp_b1, imp_w2, imp_b2, ws);
  k_gather<<<(B_ + 255) / 256, 256, 0, stream>>>(ws, label_ws, out);
}